// HHAVWGCN_84344567758926
// MI455X (gfx1250) — compile-verified
//
#include <hip/hip_runtime.h>

typedef float v2f __attribute__((ext_vector_type(2)));
typedef float v8f __attribute__((ext_vector_type(8)));

#define NNODE 4096
#define NB    32
#define CI    64
#define CO    64
#define COLS  (NB * CI)   // 2048
#define KI    192         // CHEB_K * CI
#define ED    10

// ---------------------------------------------------------------------------
// GEMM stage:  Y[b,n,c] = alpha * sum_m A[n,m] * U[b,m,c]  (+ beta * Xres[b,n,c])
// A is (N,N) row-major; U, Xres, Y are (B,N,CI).
// Logical GEMM: C[4096 x 2048] = A[4096x4096] @ Ucols[4096x2048], col = b*64+c.
// Workgroup tile 128x128, 8 waves, each wave 32x64 = 2x4 WMMA f32 16x16 tiles.
// Register double-buffered: global loads of chunk k+1 overlap WMMAs of chunk k.
// ---------------------------------------------------------------------------
#define TM   128
#define TN   128
#define TK   16
#define LSTR 20   // padded LDS row stride (words): even (b64 align), 4-mult (b128 align)

__global__ __launch_bounds__(256) void agcrn_support_gemm(
    const float* __restrict__ A, const float* __restrict__ U,
    const float* __restrict__ Xres, float* __restrict__ Y,
    float alpha, float beta)
{
    __shared__ __align__(16) float As[TM][LSTR];  // As[row][k]
    __shared__ __align__(16) float Bs[TN][LSTR];  // Bs[col][k]  (transposed stage)

    const int tid  = threadIdx.x;
    const int lane = tid & 31;
    const int wid  = tid >> 5;
    const int wm   = wid & 3;             // 4 wave-rows  of 32
    const int wn   = wid >> 2;            // 2 wave-cols  of 64
    const int rowbase = blockIdx.y * TM;
    const int colbase = blockIdx.x * TN;

    const int lr = lane & 15;             // M / Ncol within 16x16 tile
    const int lk = (lane >> 4) << 1;      // K offset {0,2} per ISA A/B layout

    v8f acc[2][4] = {};

    // global->LDS assignments (fixed per thread)
    const int ar = tid >> 1;              // A tile row 0..127
    const int ah = (tid & 1) << 3;        // 0 or 8 (half of 16-wide k chunk)
    const int bk = tid >> 4;              // U tile k 0..15
    const int bs = tid & 15;              // col sub 0..15

    // loop-invariant global bases
    const float* aRow = A + (size_t)(rowbase + ar) * NNODE + ah;
    const float* bBase[8];
#pragma unroll
    for (int i = 0; i < 8; ++i) {
        int col = colbase + bs + (i << 4);
        int b = col >> 6, c = col & 63;
        bBase[i] = U + (size_t)b * (NNODE * CI) + c;   // + k*CI per element
    }

    // ---- prologue: fetch chunk 0 into registers
    float4 aReg0 = ((const float4*)(aRow + 0))[0];
    float4 aReg1 = ((const float4*)(aRow + 0))[1];
    float  bReg[8];
#pragma unroll
    for (int i = 0; i < 8; ++i) bReg[i] = bBase[i][(size_t)bk * CI];

    for (int k0 = 0; k0 < NNODE; k0 += TK) {
        // ---- commit current chunk registers -> LDS
        *(float4*)&As[ar][ah]     = aReg0;
        *(float4*)&As[ar][ah + 4] = aReg1;
#pragma unroll
        for (int i = 0; i < 8; ++i) Bs[bs + (i << 4)][bk] = bReg[i];
        __syncthreads();

        // ---- issue next chunk's global loads (overlap with WMMAs below)
        if (k0 + TK < NNODE) {
            const float* aNext = aRow + (k0 + TK);
            aReg0 = ((const float4*)aNext)[0];
            aReg1 = ((const float4*)aNext)[1];
            const size_t krow = (size_t)(k0 + TK + bk) * CI;
#pragma unroll
            for (int i = 0; i < 8; ++i) bReg[i] = bBase[i][krow];
        }

        // ---- compute current chunk from LDS
#pragma unroll
        for (int kk = 0; kk < TK; kk += 4) {
            v2f af[2], bf[4];
#pragma unroll
            for (int mt = 0; mt < 2; ++mt)
                af[mt] = *(const v2f*)&As[wm * 32 + mt * 16 + lr][kk + lk];
#pragma unroll
            for (int nt = 0; nt < 4; ++nt)
                bf[nt] = *(const v2f*)&Bs[wn * 64 + nt * 16 + lr][kk + lk];
#pragma unroll
            for (int mt = 0; mt < 2; ++mt)
#pragma unroll
                for (int nt = 0; nt < 4; ++nt)
                    acc[mt][nt] = __builtin_amdgcn_wmma_f32_16x16x4_f32(
                        false, af[mt], false, bf[nt],
                        (short)0, acc[mt][nt], false, false);
        }
        __syncthreads();
    }

    // ---- epilogue: Y = alpha*acc (+ beta*Xres)
#pragma unroll
    for (int mt = 0; mt < 2; ++mt) {
#pragma unroll
        for (int nt = 0; nt < 4; ++nt) {
#pragma unroll
            for (int v = 0; v < 8; ++v) {
                int row = rowbase + wm * 32 + mt * 16 + v + ((lane >> 4) << 3);
                int col = colbase + wn * 64 + nt * 16 + lr;
                int b = col >> 6, c = col & 63;
                size_t idx = (size_t)b * (NNODE * CI) + (size_t)row * CI + c;
                float r = alpha * acc[mt][nt][v];
                if (beta != 0.0f) r = fmaf(beta, Xres[idx], r);
                Y[idx] = r;
            }
        }
    }
}

// ---------------------------------------------------------------------------
// Per-node adaptive projection: one workgroup per node n.
//   W_n[ki][o] = sum_d E[n,d] * Wp[d][ki][o]       (KI=192, CO=64)
//   out[b,n,o] = sum_ki XG[b][ki] * W_n[ki][o] + bias_n[o]
// XG rows: k=0 -> x, k=1 -> Y1, k=2 -> Y2.  (32x192)@(192x64) via WMMA f32.
// ---------------------------------------------------------------------------
#define PSTR 194  // even -> b64-aligned LDS rows

__global__ __launch_bounds__(256) void agcrn_node_proj(
    const float* __restrict__ x,  const float* __restrict__ y1,
    const float* __restrict__ y2, const float* __restrict__ E,
    const float* __restrict__ Wp, const float* __restrict__ biasP,
    float* __restrict__ out)
{
    __shared__ __align__(16) float WT[CO][PSTR];  // WT[o][ki]  (B operand, transposed)
    __shared__ __align__(16) float XG[NB][PSTR];  // XG[b][ki]  (A operand)
    __shared__ float ev[ED];
    __shared__ float biasS[CO];

    const int n    = blockIdx.x;
    const int tid  = threadIdx.x;
    const int lane = tid & 31;
    const int wid  = tid >> 5;

    if (tid < ED) ev[tid] = E[(size_t)n * ED + tid];
    __syncthreads();

    // node weights: 192*64 = 12288 elems, Wp reads coalesced & L2-resident
    for (int idx = tid; idx < KI * CO; idx += 256) {
        int ki = idx >> 6, o = idx & 63;
        float s = 0.0f;
#pragma unroll
        for (int d = 0; d < ED; ++d) s = fmaf(ev[d], Wp[d * (KI * CO) + idx], s);
        WT[o][ki] = s;
    }
    if (tid < CO) {
        float s = 0.0f;
#pragma unroll
        for (int d = 0; d < ED; ++d) s = fmaf(ev[d], biasP[d * CO + tid], s);
        biasS[tid] = s;
    }
    // gather x_g for this node: 32 batches x 192
    for (int idx = tid; idx < NB * KI; idx += 256) {
        int b = idx / KI, ki = idx - b * KI;
        int k = ki >> 6, i = ki & 63;
        const float* src = (k == 0) ? x : (k == 1) ? y1 : y2;
        XG[b][ki] = src[(size_t)b * (NNODE * CI) + (size_t)n * CI + i];
    }
    __syncthreads();

    // 8 waves cover the 32x64 output: wave -> (m-tile 0..1, n-tile 0..3)
    const int m0 = (wid & 1) << 4;
    const int n0 = (wid >> 1) << 4;
    const int lr = lane & 15;
    const int lk = (lane >> 4) << 1;

    v8f acc = {};
#pragma unroll
    for (int kb = 0; kb < KI; kb += 4) {
        v2f a = *(const v2f*)&XG[m0 + lr][kb + lk];
        v2f w = *(const v2f*)&WT[n0 + lr][kb + lk];
        acc = __builtin_amdgcn_wmma_f32_16x16x4_f32(
            false, a, false, w, (short)0, acc, false, false);
    }

    const int o  = n0 + lr;
    const float bia = biasS[o];
#pragma unroll
    for (int v = 0; v < 8; ++v) {
        int b = m0 + v + ((lane >> 4) << 3);
        out[(size_t)b * (NNODE * CO) + (size_t)n * CO + o] = acc[v] + bia;
    }
}

// ---------------------------------------------------------------------------
extern "C" void kernel_launch(void* const* d_in, const int* in_sizes, int n_in,
                              void* d_out, int out_size, void* d_ws, size_t ws_size,
                              hipStream_t stream) {
    const float* x     = (const float*)d_in[0];  // (32,4096,64)
    const float* E     = (const float*)d_in[1];  // (4096,10)
    const float* adj   = (const float*)d_in[2];  // (4096,4096)
    const float* Wp    = (const float*)d_in[3];  // (10,3,64,64)
    const float* biasP = (const float*)d_in[4];  // (10,64)
    float* out = (float*)d_out;

    float* y1 = (float*)d_ws;                    // 32 MB
    float* y2 = y1 + (size_t)NB * NNODE * CI;    // 32 MB

    dim3 grid(COLS / TN, NNODE / TM);            // (16, 32)
    // Y1 = A @ X
    agcrn_support_gemm<<<grid, 256, 0, stream>>>(adj, x, nullptr, y1, 1.0f, 0.0f);
    // Y2 = 2*A @ Y1 - X   (== (2A^2 - I) @ X, Chebyshev T2 without forming A^2)
    agcrn_support_gemm<<<grid, 256, 0, stream>>>(adj, y1, x, y2, 2.0f, -1.0f);
    // per-node adaptive projection + bias
    agcrn_node_proj<<<NNODE, 256, 0, stream>>>(x, y1, y2, E, Wp, biasP, out);
}